// GraphConv_85177791414564
// MI455X (gfx1250) — compile-verified
//
#include <hip/hip_runtime.h>
#include <hip/hip_bf16.h>

#define NNODE 30000
#define NF    256
#define NEDGE 960000
#define NTOT  (NNODE * NF)      // 7,680,000 floats
#define NTILE ((NNODE / 16) * (NF / 16))  // 1875 * 16 = 30000 wave-tiles

typedef float v2f __attribute__((ext_vector_type(2)));
typedef float v8f __attribute__((ext_vector_type(8)));

// ---------------- utility ----------------
__global__ void zero_f32(float* __restrict__ p, int n) {
  int i = blockIdx.x * blockDim.x + threadIdx.x;
  if (i < n) p[i] = 0.0f;
}

// deg[col[e]] += ew[e]   (self-loop +1 folded into dinv_kernel)
__global__ void degree_kernel(const int* __restrict__ col,
                              const float* __restrict__ ew,
                              float* __restrict__ deg, int e) {
  int i = blockIdx.x * blockDim.x + threadIdx.x;
  if (i < e) unsafeAtomicAdd(&deg[col[i]], ew[i]);
}

// deg -> dinv = rsqrt(deg + 1) in place
__global__ void dinv_kernel(float* __restrict__ deg, int n) {
  int i = blockIdx.x * blockDim.x + threadIdx.x;
  if (i < n) {
    float d = deg[i] + 1.0f;              // +1 = self-loop weight
    deg[i] = (d > 0.0f) ? rsqrtf(d) : 0.0f;
  }
}

// norm[e] = dinv[row]*ew*dinv[col]
__global__ void norm_kernel(const int* __restrict__ row, const int* __restrict__ col,
                            const float* __restrict__ ew, const float* __restrict__ dinv,
                            float* __restrict__ norm, int e) {
  int i = blockIdx.x * blockDim.x + threadIdx.x;
  if (i < e) norm[i] = dinv[row[i]] * ew[i] * dinv[col[i]];
}

// Y[i,:] = dinv[i]^2 * X[i,:]   (self-loop contribution doubles as the zero-init)
__global__ void self_init(const float* __restrict__ dinv,
                          const float* __restrict__ X, float* __restrict__ Y) {
  int i = blockIdx.x * blockDim.x + threadIdx.x;   // float4 index
  if (i >= NTOT / 4) return;
  int node = i >> 6;                               // 64 float4 per node (F=256)
  float di = dinv[node];
  float w  = di * di;
  float4 v = ((const float4*)X)[i];
  float4 o;
  o.x = w * v.x; o.y = w * v.y; o.z = w * v.z; o.w = w * v.w;
  ((float4*)Y)[i] = o;
}

// scatter-add: Y[col[e],:] += norm[e] * X[row[e],:] ; 64 lanes/edge, float4 each.
// Working set (~80 MB) fits MI455X's 192 MB L2 -> gathers & fp32 atomics stay in L2.
__global__ void spmm_edges(const int* __restrict__ row, const int* __restrict__ col,
                           const float* __restrict__ norm,
                           const float* __restrict__ X, float* __restrict__ Y) {
  long long t = (long long)blockIdx.x * blockDim.x + threadIdx.x;
  int e  = (int)(t >> 6);
  if (e >= NEDGE) return;
  int f4 = ((int)t & 63) << 2;
  int r = row[e], c = col[e];
  float w = norm[e];
  const float4 v = *(const float4*)(X + (size_t)r * NF + f4);
  float* dst = Y + (size_t)c * NF + f4;
  unsafeAtomicAdd(dst + 0, w * v.x);
  unsafeAtomicAdd(dst + 1, w * v.y);
  unsafeAtomicAdd(dst + 2, w * v.z);
  unsafeAtomicAdd(dst + 3, w * v.w);
}

// Z = prelu(X @ W^T + b); per-column sum/sumsq accumulated into stats[0:256]/[256:512].
// One wave per 16x16 output tile, fp32 WMMA (V_WMMA_F32_16X16X4_F32), K loop of 64.
__global__ void gemm_prelu_stats(const float* __restrict__ X,   // [NNODE, NF]
                                 const float* __restrict__ W,   // [NF, NF], out=X@W^T
                                 const float* __restrict__ bias,
                                 const float* __restrict__ pw,
                                 float* __restrict__ Z,
                                 float* __restrict__ stats) {
  int wave = (blockIdx.x * blockDim.x + threadIdx.x) >> 5;
  int lane = threadIdx.x & 31;
  if (wave >= NTILE) return;                 // uniform per wave; grid sized exactly
  int tm   = wave / (NF / 16);               // 0..1874
  int tn   = wave % (NF / 16);               // 0..15
  int half = lane >> 4;                      // 0: K pair {0,1}; 1: K pair {2,3}
  int l16  = lane & 15;

  const float* arow = X + (size_t)(tm * 16 + l16) * NF;  // A: M = l16
  const float* brow = W + (size_t)(tn * 16 + l16) * NF;  // B[k][n] = W[n][k] -> N = l16
  v8f c = {};
  for (int k0 = 0; k0 < NF; k0 += 4) {
    int kb = k0 + 2 * half;
    v2f a, b;
    float2 av = *(const float2*)(arow + kb);
    float2 bv = *(const float2*)(brow + kb);
    a.x = av.x; a.y = av.y;
    b.x = bv.x; b.y = bv.y;
    c = __builtin_amdgcn_wmma_f32_16x16x4_f32(false, a, false, b,
                                              (short)0, c, false, false);
  }

  int colg = tn * 16 + l16;
  float bb = bias[colg];
  float pv = pw[colg];
  float s = 0.0f, ss = 0.0f;
#pragma unroll
  for (int r = 0; r < 8; ++r) {
    int rowg = tm * 16 + r + 8 * half;       // C layout: lane>=16 carries rows 8..15
    float v = c[r] + bb;
    v = (v >= 0.0f) ? v : pv * v;
    Z[(size_t)rowg * NF + colg] = v;
    s += v; ss += v * v;
  }
  unsafeAtomicAdd(&stats[colg], s);
  unsafeAtomicAdd(&stats[NF + colg], ss);
}

// stats(sum,sumsq) -> (scale, shift) with gamma/beta folded in
__global__ void bn_finalize(float* __restrict__ stats,
                            const float* __restrict__ gamma,
                            const float* __restrict__ beta) {
  int f = threadIdx.x;
  if (f >= NF) return;
  const float invn = 1.0f / (float)NNODE;
  float mean = stats[f] * invn;
  float var  = stats[NF + f] * invn - mean * mean;   // biased, training-mode BN
  float scale = gamma[f] * rsqrtf(var + 1e-5f);
  stats[f]      = scale;
  stats[NF + f] = beta[f] - mean * scale;
}

// out = Z * scale[f] + shift[f]
__global__ void bn_apply(const float* __restrict__ Z, const float* __restrict__ stats,
                         float* __restrict__ out) {
  int i = blockIdx.x * blockDim.x + threadIdx.x;    // float4 index
  if (i >= NTOT / 4) return;
  int f4 = (i & 63) << 2;
  float4 v  = ((const float4*)Z)[i];
  float4 sc = *(const float4*)(stats + f4);
  float4 sh = *(const float4*)(stats + NF + f4);
  float4 o;
  o.x = v.x * sc.x + sh.x; o.y = v.y * sc.y + sh.y;
  o.z = v.z * sc.z + sh.z; o.w = v.w * sc.w + sh.w;
  ((float4*)out)[i] = o;
}

extern "C" void kernel_launch(void* const* d_in, const int* in_sizes, int n_in,
                              void* d_out, int out_size, void* d_ws, size_t ws_size,
                              hipStream_t stream) {
  const float* cell  = (const float*)d_in[0];
  const float* sub   = (const float*)d_in[1];
  const int*   ei    = (const int*)d_in[2];
  const float* ew    = (const float*)d_in[3];
  const float* W1    = (const float*)d_in[4];
  const float* b1    = (const float*)d_in[5];
  const float* W2    = (const float*)d_in[6];
  const float* b2    = (const float*)d_in[7];
  const float* pw    = (const float*)d_in[8];
  const float* gamma = (const float*)d_in[9];
  const float* beta  = (const float*)d_in[10];
  float* out = (float*)d_out;

  float* bufA  = (float*)d_ws;            // 30.72 MB
  float* bufB  = bufA + NTOT;             // 30.72 MB
  float* dinv  = bufB + NTOT;             // 120 KB
  float* norm  = dinv + NNODE;            // 3.84 MB
  float* stats = norm + NEDGE;            // 2 KB
  const int* row = ei;
  const int* col = ei + NEDGE;

  // x = concat(cell_feat, sub_feat)
  hipMemcpyAsync(bufA, cell, (size_t)66 * NF * sizeof(float),
                 hipMemcpyDeviceToDevice, stream);
  hipMemcpyAsync(bufA + 66 * NF, sub, (size_t)29934 * NF * sizeof(float),
                 hipMemcpyDeviceToDevice, stream);

  // gcn_norm
  zero_f32<<<(NNODE + 255) / 256, 256, 0, stream>>>(dinv, NNODE);
  degree_kernel<<<(NEDGE + 255) / 256, 256, 0, stream>>>(col, ew, dinv, NEDGE);
  dinv_kernel<<<(NNODE + 255) / 256, 256, 0, stream>>>(dinv, NNODE);
  norm_kernel<<<(NEDGE + 255) / 256, 256, 0, stream>>>(row, col, ew, dinv, norm, NEDGE);

  const int elem4Blocks = (NTOT / 4 + 255) / 256;
  const long long edgeThreads = (long long)NEDGE * 64;
  const int edgeBlocks = (int)((edgeThreads + 255) / 256);

  float* cur = bufA;
  float* oth = bufB;
  for (int layer = 0; layer < 2; ++layer) {
    const float* W  = layer ? W2 : W1;
    const float* bs = layer ? b2 : b1;

    // K=3 hops of x <- A_hat x
    for (int h = 0; h < 3; ++h) {
      self_init<<<elem4Blocks, 256, 0, stream>>>(dinv, cur, oth);
      spmm_edges<<<edgeBlocks, 256, 0, stream>>>(row, col, norm, cur, oth);
      float* t = cur; cur = oth; oth = t;
    }

    // linear + PReLU + BN-stat accumulation (WMMA fp32)
    zero_f32<<<2, 256, 0, stream>>>(stats, 2 * NF);
    gemm_prelu_stats<<<NTILE / 8, 256, 0, stream>>>(cur, W, bs, pw, oth, stats);
    { float* t = cur; cur = oth; oth = t; }   // cur = pre-BN activations

    bn_finalize<<<1, 256, 0, stream>>>(stats, gamma, beta);
    float* dst = (layer == 1) ? out : oth;
    bn_apply<<<elem4Blocks, 256, 0, stream>>>(cur, stats, dst);
    if (layer == 0) { float* t = cur; cur = oth; oth = t; }  // cur = BN output
  }
}